// PrototypeLoss_78615081386035
// MI455X (gfx1250) — compile-verified
//
#include <hip/hip_runtime.h>
#include <hip/hip_bf16.h>
#include <float.h>

// ---------------- problem constants ----------------
#define P_COLS        1000
#define F4_PER_ROW    250            // 1000 floats / 4
#define FULL_CHUNKS   7              // 7 * 32 lanes = 224 float4 staged async
#define TAIL_F4       26             // 250 - 224 handled by direct loads
#define WAVES_PER_BLK 4
#define BLK_THREADS   (WAVES_PER_BLK * 32)
#define N_BUF         4              // async pipeline depth (3 chunks in flight)

// ---------------- gfx1250 async global->LDS path ----------------
#if defined(__has_builtin)
#  if __has_builtin(__builtin_amdgcn_global_load_async_to_lds_b128)
#    define ASYNC_LDS_PATH 1
#  endif
#  if __has_builtin(__builtin_amdgcn_s_wait_asynccnt)
#    define HAVE_WAIT_ASYNC_BUILTIN 1
#  endif
#endif

#if !defined(ASYNC_LDS_PATH)
#warning "gfx1250 async-to-LDS builtin NOT available; using synchronous fallback"
#endif

// Builtin signature (probe-confirmed via diagnostic): 16-byte int vector
// pointers in addrspace(1) (global) / addrspace(3) (LDS), imm offset, imm cpol.
typedef int v4i __attribute__((vector_size(4 * sizeof(int))));
typedef __attribute__((address_space(1))) v4i g_v4i;
typedef __attribute__((address_space(3))) v4i l_v4i;

__device__ __forceinline__ void async_copy_f4(const float4* g, float4* l) {
#if defined(ASYNC_LDS_PATH)
    __builtin_amdgcn_global_load_async_to_lds_b128(
        (g_v4i*)g, (l_v4i*)l, /*imm offset*/0, /*cpol*/0);
#else
    *l = *g;   // synchronous fallback (still correct)
#endif
}

template <int N>
__device__ __forceinline__ void wait_async() {
#if defined(ASYNC_LDS_PATH)
#  if defined(HAVE_WAIT_ASYNC_BUILTIN)
    __builtin_amdgcn_s_wait_asynccnt(N);
#  else
    asm volatile("s_wait_asynccnt %0" ::"i"(N) : "memory");
#  endif
#endif
}

// One pipeline step: issue async prefetch of chunk C+3 (if any), wait until
// chunk C has landed in LDS (<= W outstanding), consume it.
template <int C, int W>
__device__ __forceinline__ void pipe_step(const float4* __restrict__ src,
                                          float4 (&wbuf)[N_BUF][32],
                                          int lane, float& vsum, float& vmin)
{
    if ((C + N_BUF - 1) < FULL_CHUNKS) {
        async_copy_f4(&src[(C + N_BUF - 1) * 32 + lane],
                      &wbuf[(C + N_BUF - 1) & (N_BUF - 1)][lane]);
    }
    wait_async<W>();
    float4 v = wbuf[C & (N_BUF - 1)][lane];          // ds_load_b128
    vsum += (v.x + v.y) + (v.z + v.w);
    vmin  = fminf(vmin, fminf(fminf(v.x, v.y), fminf(v.z, v.w)));
}

// ---------------- kernel 1: per-row (min, sum, pick) ----------------
// One wave32 per row. 4-deep async global->LDS pipeline (3 x 512B chunks in
// flight per wave) + hoisted tail/gather loads, then ds_load_b128 + VALU
// min/sum, then wave32 shfl_xor reduction.
__global__ void __launch_bounds__(BLK_THREADS)
row_stats_kernel(const float* __restrict__ d,
                 const int*   __restrict__ labels,
                 float*       __restrict__ row_out)
{
    __shared__ float4 buf[WAVES_PER_BLK][N_BUF][32];

    const int lane = threadIdx.x & 31;
    const int wave = threadIdx.x >> 5;
    const int row  = blockIdx.x * WAVES_PER_BLK + wave;

    const float*  rowp = d + (size_t)row * P_COLS;
    const float4* src  = (const float4*)rowp;     // 4000B rows -> 16B aligned

    // ---- prologue: fill the async pipeline (chunks 0..2 in flight) ----
    async_copy_f4(&src[0 * 32 + lane], &buf[wave][0][lane]);
    async_copy_f4(&src[1 * 32 + lane], &buf[wave][1][lane]);
    async_copy_f4(&src[2 * 32 + lane], &buf[wave][2][lane]);

    // ---- hoisted independent loads: tail chunk + in-class gather ----
    float4 tv = make_float4(0.0f, 0.0f, 0.0f, 0.0f);
    const bool has_tail = (lane < TAIL_F4);
    if (has_tail) tv = src[FULL_CHUNKS * 32 + lane];      // global_load_b128
    const int   lbl      = labels[row];                    // wave-uniform
    const float in_class = rowp[lbl];                      // dependent gather

    float vmin = FLT_MAX;
    float vsum = 0.0f;

    // Explicitly unrolled: steady state keeps <=3 async ops outstanding,
    // then drains with 2/1/0 (s_wait_asynccnt needs literal immediates).
    pipe_step<0, 3>(src, buf[wave], lane, vsum, vmin);
    pipe_step<1, 3>(src, buf[wave], lane, vsum, vmin);
    pipe_step<2, 3>(src, buf[wave], lane, vsum, vmin);
    pipe_step<3, 3>(src, buf[wave], lane, vsum, vmin);
    pipe_step<4, 2>(src, buf[wave], lane, vsum, vmin);
    pipe_step<5, 1>(src, buf[wave], lane, vsum, vmin);
    pipe_step<6, 0>(src, buf[wave], lane, vsum, vmin);

    // tail: float4 indices [224, 250), lanes 0..25
    if (has_tail) {
        vsum += (tv.x + tv.y) + (tv.z + tv.w);
        vmin  = fminf(vmin, fminf(fminf(tv.x, tv.y), fminf(tv.z, tv.w)));
    }

    // wave32 reduction (5 steps)
    #pragma unroll
    for (int off = 16; off > 0; off >>= 1) {
        vsum += __shfl_xor(vsum, off, 32);
        vmin  = fminf(vmin, __shfl_xor(vmin, off, 32));
    }

    if (lane == 0) {
        const float denom = vsum - (float)P_COLS * vmin;   // sum(d - min)
        row_out[row] = (in_class - vmin) / denom;
    }
}

// ---------------- kernel 2: deterministic mean over B rows ----------------
__global__ void __launch_bounds__(256)
reduce_mean_kernel(const float* __restrict__ row_vals,
                   float*       __restrict__ out,
                   int nrows)
{
    __shared__ float s[256];
    float acc = 0.0f;
    // fixed per-thread visitation order -> bit-deterministic across replays
    for (int i = threadIdx.x; i < nrows; i += 256)
        acc += row_vals[i];
    s[threadIdx.x] = acc;
    __syncthreads();
    #pragma unroll
    for (int stride = 128; stride > 0; stride >>= 1) {
        if (threadIdx.x < stride) s[threadIdx.x] += s[threadIdx.x + stride];
        __syncthreads();
    }
    if (threadIdx.x == 0)
        out[0] = s[0] / (float)nrows;
}

// ---------------- launch ----------------
extern "C" void kernel_launch(void* const* d_in, const int* in_sizes, int n_in,
                              void* d_out, int out_size, void* d_ws, size_t ws_size,
                              hipStream_t stream)
{
    const float* d      = (const float*)d_in[0];   // [B, 1000] float32
    const int*   labels = (const int*)d_in[1];     // [B] integer
    float*       out    = (float*)d_out;           // scalar float32
    float*       row_ws = (float*)d_ws;            // B floats of scratch

    const int B = in_sizes[1];                     // 65536

    row_stats_kernel<<<B / WAVES_PER_BLK, BLK_THREADS, 0, stream>>>(d, labels, row_ws);
    reduce_mean_kernel<<<1, 256, 0, stream>>>(row_ws, out, B);
}